// down_block_47845935677648
// MI455X (gfx1250) — compile-verified
//
#include <hip/hip_runtime.h>

#define N_IN    163842
#define N_OUTN  40962
#define IN_CH   64
#define OUT_CH  128
#define KC1     (7 * IN_CH)    // 448
#define KC2     (7 * OUT_CH)   // 896
#define KP1     (KC1 + 4)      // padded LDS stride (16B-aligned rows, bank spread)
#define KP2     (KC2 + 4)

typedef __attribute__((ext_vector_type(2))) float v2f;
typedef __attribute__((ext_vector_type(4))) float v4f;
typedef __attribute__((ext_vector_type(8))) float v8f;

// ---------------------------------------------------------------------------
// 0) Pack W (K x 128, row-major) into the WMMA B-fragment layout:
//    Wp[((nt*(K/8) + kk)*32 + lane)*4 + j], so one global_load_b128 per lane
//    supplies the B fragments for two consecutive 16x16x4 WMMA steps.
//    lane = 16*h + l16;  j=0,1 -> W[8kk+2h + j][nt*16+l16]
//                        j=2,3 -> W[8kk+4+2h + (j-2)][nt*16+l16]
// ---------------------------------------------------------------------------
template <int K>
__global__ void packW_kernel(const float* __restrict__ W, float* __restrict__ Wp) {
  const int e = blockIdx.x * 256 + threadIdx.x;
  if (e >= K * OUT_CH) return;
  int t = e;
  const int j = t & 3;   t >>= 2;
  const int lane = t & 31; t >>= 5;
  const int kk = t % (K / 8);
  const int nt = t / (K / 8);
  const int h = lane >> 4;
  const int l16 = lane & 15;
  const int row = 8 * kk + 2 * h + (j & 1) + 4 * (j >> 1);
  const int col = nt * 16 + l16;
  Wp[e] = W[row * OUT_CH + col];
}

// ---------------------------------------------------------------------------
// 1) Mean-pool over the 1-ring with the reference's (N, feat, 7) reshape quirk
// ---------------------------------------------------------------------------
__global__ void pool_kernel(const float* __restrict__ x,
                            const int* __restrict__ pno,
                            float* __restrict__ pooled) {
  __shared__ float lds[4 * 7 * IN_CH];       // 4 nodes * 448 floats
  const int node0 = blockIdx.x * 4;
  const int tid = threadIdx.x;               // 256 threads

  for (int t = tid; t < 4 * 448; t += 256) {
    const int ln = t / 448;
    const int i  = t - ln * 448;
    const int r  = i >> 6;
    const int c  = i & 63;
    const int node = node0 + ln;
    float v = 0.0f;
    if (node < N_OUTN) {
      const int src = pno[node * 7 + r];
      v = x[src * IN_CH + c];
    }
    lds[ln * 448 + i] = v;
  }
  __syncthreads();

  for (int t = tid; t < 4 * IN_CH; t += 256) {
    const int ln = t >> 6;
    const int c  = t & 63;
    const int node = node0 + ln;
    if (node < N_OUTN) {
      const float* f = &lds[ln * 448 + c * 7];
      const float s = f[0] + f[1] + f[2] + f[3] + f[4] + f[5] + f[6];
      pooled[node * IN_CH + c] = s * (1.0f / 7.0f);
    }
  }
}

// ---------------------------------------------------------------------------
// 2) Gather + GEMM via V_WMMA_F32_16X16X4_F32.
//    256 threads = 8 waves; wave w owns N-tile [16w,16w+16).
//    TWO 16-row M-tiles per block (32 rows) so each B fragment feeds 2 WMMAs.
//    A panel (32 x K) staged in dynamic LDS (padded stride KP).
//    Optionally fuses BN+LeakyReLU of the input rows into the gather.
// ---------------------------------------------------------------------------
template <int K, int KP, int C>
__global__ void gemm_kernel(const float* __restrict__ Ain,   // (rows, C)
                            const int* __restrict__ no,      // (N_OUTN*7)
                            const float* __restrict__ Wp,    // packed (K*128)
                            const float* __restrict__ bias,  // (128)
                            const float* __restrict__ scale, // (C)
                            const float* __restrict__ shift, // (C)
                            int fuse_bn,
                            float* __restrict__ out)         // (N_OUTN, 128)
{
  extern __shared__ float ldsA[];            // 32 * KP floats
  const int m0  = blockIdx.x * 32;
  const int tid = threadIdx.x;

  // cooperative vectorized gather: A[m][j*C + c..c+3] = Ain[no[(m0+m)*7+j], c..c+3]
  for (int t = tid; t < 32 * (K / 4); t += 256) {
    const int m   = t / (K / 4);
    const int rem = (t - m * (K / 4)) * 4;
    const int j   = rem / C;
    const int c   = rem - j * C;
    int row = m0 + m;
    if (row >= N_OUTN) row = N_OUTN - 1;     // safe clamp; stores are guarded
    const int src = no[row * 7 + j];
    v4f v = *(const v4f*)&Ain[src * C + c];
    if (fuse_bn) {
#pragma unroll
      for (int q = 0; q < 4; ++q) {
        float u = v[q] * scale[c + q] + shift[c + q];
        v[q] = (u >= 0.0f) ? u : 0.2f * u;
      }
    }
    *(v4f*)&ldsA[m * KP + rem] = v;
  }
  __syncthreads();

  const int wave = tid >> 5;                 // 0..7 -> N-tile
  const int lane = tid & 31;
  const int half = lane >> 4;
  const int l16  = lane & 15;
  const int ncol = wave * 16 + l16;

  const float* ap0 = &ldsA[l16 * KP + 2 * half];
  const float* ap1 = ap0 + 16 * KP;
  const float* wq  = Wp + wave * (K / 8) * 128 + lane * 4;

  v8f acc0 = {}, acc1 = {};
#pragma unroll 2
  for (int k = 0; k < K; k += 8) {
    const v4f bq = *(const v4f*)(wq + (k >> 3) * 128);  // one b128: B for 2 steps
    v2f b0; b0.x = bq.x; b0.y = bq.y;
    v2f b1; b1.x = bq.z; b1.y = bq.w;
    const v2f a00 = *(const v2f*)(ap0 + k);
    const v2f a01 = *(const v2f*)(ap0 + k + 4);
    const v2f a10 = *(const v2f*)(ap1 + k);
    const v2f a11 = *(const v2f*)(ap1 + k + 4);
    acc0 = __builtin_amdgcn_wmma_f32_16x16x4_f32(false, a00, false, b0, (short)0, acc0, false, false);
    acc1 = __builtin_amdgcn_wmma_f32_16x16x4_f32(false, a10, false, b0, (short)0, acc1, false, false);
    acc0 = __builtin_amdgcn_wmma_f32_16x16x4_f32(false, a01, false, b1, (short)0, acc0, false, false);
    acc1 = __builtin_amdgcn_wmma_f32_16x16x4_f32(false, a11, false, b1, (short)0, acc1, false, false);
  }

  const float bv = bias[ncol];
  const int rbase = m0 + 8 * half;           // C/D layout: VGPR v -> M = v + 8*half
  if (m0 + 32 <= N_OUTN) {                   // fast path: full tile, no masking
#pragma unroll
    for (int v = 0; v < 8; ++v) {
      out[(rbase + v) * OUT_CH + ncol]      = acc0[v] + bv;
      out[(rbase + 16 + v) * OUT_CH + ncol] = acc1[v] + bv;
    }
  } else {
#pragma unroll
    for (int v = 0; v < 8; ++v) {
      if (rbase + v < N_OUTN)      out[(rbase + v) * OUT_CH + ncol]      = acc0[v] + bv;
      if (rbase + 16 + v < N_OUTN) out[(rbase + 16 + v) * OUT_CH + ncol] = acc1[v] + bv;
    }
  }
}

// ---------------------------------------------------------------------------
// 3) Deterministic two-stage per-column mean/var (biased) for BatchNorm
// ---------------------------------------------------------------------------
#define CHUNK_ROWS 512

__global__ void colstats_kernel(const float* __restrict__ y,
                                float* __restrict__ partials, int nrows) {
  const int c = threadIdx.x;                 // 128 threads = 128 columns
  const int chunk = blockIdx.x;
  const int r0 = chunk * CHUNK_ROWS;
  int r1 = r0 + CHUNK_ROWS;
  if (r1 > nrows) r1 = nrows;
  float s = 0.0f, s2 = 0.0f;
  for (int r = r0; r < r1; ++r) {
    const float v = y[r * OUT_CH + c];
    s += v;
    s2 += v * v;
  }
  partials[chunk * 256 + c] = s;
  partials[chunk * 256 + 128 + c] = s2;
}

__global__ void bnfinal_kernel(const float* __restrict__ partials, int nchunks,
                               const float* __restrict__ gamma,
                               const float* __restrict__ beta,
                               float* __restrict__ scale,
                               float* __restrict__ shift, int nrows) {
  const int c = threadIdx.x;                 // 128 threads
  float s = 0.0f, s2 = 0.0f;
  for (int i = 0; i < nchunks; ++i) {        // fixed order -> deterministic
    s += partials[i * 256 + c];
    s2 += partials[i * 256 + 128 + c];
  }
  const float inv = 1.0f / (float)nrows;
  const float m = s * inv;
  float var = s2 * inv - m * m;
  if (var < 0.0f) var = 0.0f;
  const float sc = gamma[c] * rsqrtf(var + 1e-5f);
  scale[c] = sc;
  shift[c] = beta[c] - m * sc;
}

// ---------------------------------------------------------------------------
// 4) Final BN + LeakyReLU -> d_out
// ---------------------------------------------------------------------------
__global__ void apply_kernel(const float* __restrict__ y,
                             const float* __restrict__ scale,
                             const float* __restrict__ shift,
                             float* __restrict__ out, int n) {
  const int i = blockIdx.x * 256 + threadIdx.x;
  if (i < n) {
    const int c = i & (OUT_CH - 1);
    float v = y[i] * scale[c] + shift[c];
    out[i] = (v >= 0.0f) ? v : 0.2f * v;
  }
}

// ---------------------------------------------------------------------------
extern "C" void kernel_launch(void* const* d_in, const int* in_sizes, int n_in,
                              void* d_out, int out_size, void* d_ws, size_t ws_size,
                              hipStream_t stream) {
  const float* x   = (const float*)d_in[0];
  const int*   no  = (const int*)d_in[1];    // neigh_orders  (values < N_OUTN)
  const int*   pno = (const int*)d_in[2];    // pool_neigh_orders (values < N_IN)
  const float* W1  = (const float*)d_in[3];
  const float* b1  = (const float*)d_in[4];
  const float* g1  = (const float*)d_in[5];
  const float* be1 = (const float*)d_in[6];
  const float* W2  = (const float*)d_in[7];
  const float* b2  = (const float*)d_in[8];
  const float* g2  = (const float*)d_in[9];
  const float* be2 = (const float*)d_in[10];
  float* out = (float*)d_out;

  char* ws = (char*)d_ws;
  float* pooled   = (float*)(ws + 0);                      // 40962*64  f32
  float* y1       = (float*)(ws + 10486272);               // 40962*128 f32
  float* y2       = (float*)(ws + 31458816);               // 40962*128 f32
  float* partials = (float*)(ws + 52431360);               // <=81*256  f32
  float* scale1   = (float*)(ws + 52514304);
  float* shift1   = scale1 + 128;
  float* scale2   = shift1 + 128;
  float* shift2   = scale2 + 128;
  float* Wp1      = (float*)(ws + 52516352);               // 448*128 f32
  float* Wp2      = (float*)(ws + 52745728);               // 896*128 f32

  const int nchunks = (N_OUTN + CHUNK_ROWS - 1) / CHUNK_ROWS;   // 81
  const int mblocks = (N_OUTN + 31) / 32;                       // 1281

  packW_kernel<KC1><<<(KC1 * OUT_CH + 255) / 256, 256, 0, stream>>>(W1, Wp1);
  packW_kernel<KC2><<<(KC2 * OUT_CH + 255) / 256, 256, 0, stream>>>(W2, Wp2);

  pool_kernel<<<(N_OUTN + 3) / 4, 256, 0, stream>>>(x, pno, pooled);

  gemm_kernel<KC1, KP1, IN_CH>
      <<<mblocks, 256, 32 * KP1 * sizeof(float), stream>>>(
          pooled, no, Wp1, b1, scale1, shift1, /*fuse_bn=*/0, y1);

  colstats_kernel<<<nchunks, 128, 0, stream>>>(y1, partials, N_OUTN);
  bnfinal_kernel<<<1, 128, 0, stream>>>(partials, nchunks, g1, be1,
                                        scale1, shift1, N_OUTN);

  gemm_kernel<KC2, KP2, OUT_CH>
      <<<mblocks, 256, 32 * KP2 * sizeof(float), stream>>>(
          y1, no, Wp2, b2, scale1, shift1, /*fuse_bn=*/1, y2);

  colstats_kernel<<<nchunks, 128, 0, stream>>>(y2, partials, N_OUTN);
  bnfinal_kernel<<<1, 128, 0, stream>>>(partials, nchunks, g2, be2,
                                        scale2, shift2, N_OUTN);

  apply_kernel<<<(N_OUTN * OUT_CH + 255) / 256, 256, 0, stream>>>(
      y2, scale2, shift2, out, N_OUTN * OUT_CH);
}